// MV_GNN_Graph_41618233099040
// MI455X (gfx1250) — compile-verified
//
#include <hip/hip_runtime.h>

// ---------------------------------------------------------------------------
// MV_GNN on MI455X (gfx1250, wave32).
// Algebraic decomposition: segment_sum(concat[h[dst],h[src],ea], dst) =
//   [deg*h , A@h , S_ea]  with S_ea layer-invariant. Dense 64-wide GEMMs use
//   V_WMMA_F32_16X16X4_F32 (full fp32 matrix cores); SpMM via f32 atomics.
// Weights staged TRANSPOSED in LDS so each B fragment is one ds_load_b64.
// ---------------------------------------------------------------------------

typedef __attribute__((ext_vector_type(2))) float v2f;
typedef __attribute__((ext_vector_type(8))) float v8f;

#define NN 50000      // nodes
#define NE 800000     // edges (without self loops)
#define NGR 256       // graphs
#define NF 64         // node feature dim
#define EF 16         // edge feature dim
#define HD 64         // hidden dim
#define TILES (NN/16) // 3125 exactly -> EXEC all-ones in every WMMA wave
#define WPB 8         // waves per block (256 threads)

__device__ __forceinline__ v8f wmma4(v2f a, v2f b, v8f c) {
  // D = A(16x4,f32) x B(4x16,f32) + C(16x16,f32)
  return __builtin_amdgcn_wmma_f32_16x16x4_f32(
      /*neg_a=*/false, a, /*neg_b=*/false, b,
      /*c_mod=*/(short)0, c, /*reuse_a=*/false, /*reuse_b=*/false);
}

__device__ __forceinline__ v8f vzero8() {
  v8f z;
#pragma unroll
  for (int i = 0; i < 8; ++i) z[i] = 0.0f;
  return z;
}

// Stage W[K][64] (row-major global) into LDS transposed: sWt[n*K + k].
template <int KD>
__device__ __forceinline__ void stage_wt(const float* __restrict__ W,
                                         float* __restrict__ sWt) {
  for (int idx = threadIdx.x; idx < KD * 64; idx += blockDim.x) {
    int k = idx >> 6, n = idx & 63;      // coalesced global read
    sWt[n * KD + k] = W[idx];            // strided LDS write (one-time)
  }
}

// --- self-loop init: Ax = x, S_ea = 1, deg = 1 ------------------------------
__global__ void init_node_kernel(const float* __restrict__ x,
                                 float* __restrict__ Ax,
                                 float* __restrict__ S_ea,
                                 float* __restrict__ deg) {
  int t = blockIdx.x * blockDim.x + threadIdx.x;
  if (t >= NN * NF) return;
  int i = t >> 6, f = t & 63;
  Ax[t] = x[t];
  if (f < EF) S_ea[i * EF + f] = 1.0f;
  if (f == 0) deg[i] = 1.0f;
}

// --- edge pass for the layer-invariant parts: Ax += x[src], S_ea += ea, deg++
__global__ void edge_accum_x_kernel(const float* __restrict__ x,
                                    const float* __restrict__ ea,
                                    const int* __restrict__ ei,
                                    float* __restrict__ Ax,
                                    float* __restrict__ S_ea,
                                    float* __restrict__ deg) {
  long long t = (long long)blockIdx.x * blockDim.x + threadIdx.x;
  if (t >= (long long)NE * 64) return;
  int e = (int)(t >> 6), f = (int)(t & 63);
  int s = ei[e];        // edge_index[0] : source (x_j)
  int d = ei[NE + e];   // edge_index[1] : target (x_i / dst)
  atomicAdd(&Ax[d * NF + f], x[s * NF + f]);
  if (f < EF) atomicAdd(&S_ea[d * EF + f], ea[e * EF + f]);
  if (f == 0) atomicAdd(&deg[d], 1.0f);
}

// --- per-layer edge pass: Ah += h[src] (Ah pre-seeded with h = self loop) ---
__global__ void edge_accum_h_kernel(const float* __restrict__ h,
                                    const int* __restrict__ ei,
                                    float* __restrict__ Ah) {
  long long t = (long long)blockIdx.x * blockDim.x + threadIdx.x;
  if (t >= (long long)NE * 64) return;
  int e = (int)(t >> 6), f = (int)(t & 63);
  int s = ei[e];
  int d = ei[NE + e];
  atomicAdd(&Ah[d * HD + f], h[s * HD + f]);
}

// --- h0 = relu(x@W_nh + b_nh);  c = h0 + b_mh + S_ea @ W_mh[128:144] --------
__global__ void h0_c_kernel(const float* __restrict__ x,
                            const float* __restrict__ W_nh,
                            const float* __restrict__ b_nh,
                            const float* __restrict__ W_mh,
                            const float* __restrict__ b_mh,
                            const float* __restrict__ S_ea,
                            float* __restrict__ h0,
                            float* __restrict__ cc) {
  __shared__ __align__(16) float sWnh[64 * NF];   // 16 KB, [n][k]
  __shared__ __align__(16) float sW3[64 * EF];    // 4 KB,  [n][k] (W_mh rows 128..143)
  __shared__ float sBn[HD];
  __shared__ float sBm[HD];
  stage_wt<NF>(W_nh, sWnh);
  stage_wt<EF>(W_mh + 128 * HD, sW3);
  if (threadIdx.x < HD) { sBn[threadIdx.x] = b_nh[threadIdx.x];
                          sBm[threadIdx.x] = b_mh[threadIdx.x]; }
  __syncthreads();

  int lane = threadIdx.x & 31, wave = threadIdx.x >> 5;
  int tile = blockIdx.x * WPB + wave;
  if (tile >= TILES) return;                 // uniform per wave -> EXEC stays full
  int m = lane & 15, kh = lane >> 4;
  int row = tile * 16 + m;

  v8f acc[4], acc2[4];
#pragma unroll
  for (int nt = 0; nt < 4; ++nt) { acc[nt] = vzero8(); acc2[nt] = vzero8(); }

  // x @ W_nh  (K = 64)
#pragma unroll 4
  for (int ks = 0; ks < 16; ++ks) {
    int k = ks * 4 + kh * 2;
    v2f a = *(const v2f*)&x[row * NF + k];
#pragma unroll
    for (int nt = 0; nt < 4; ++nt) {
      v2f b = *(const v2f*)&sWnh[(nt * 16 + m) * NF + k];   // one ds_load_b64
      acc[nt] = wmma4(a, b, acc[nt]);
    }
  }
  // S_ea @ W3  (K = 16)
#pragma unroll
  for (int ks = 0; ks < 4; ++ks) {
    int k = ks * 4 + kh * 2;
    v2f a = *(const v2f*)&S_ea[row * EF + k];
#pragma unroll
    for (int nt = 0; nt < 4; ++nt) {
      v2f b = *(const v2f*)&sW3[(nt * 16 + m) * EF + k];
      acc2[nt] = wmma4(a, b, acc2[nt]);
    }
  }
  // C/D layout: vgpr v, lane l -> row v + 8*(l>>4), col l&15
#pragma unroll
  for (int nt = 0; nt < 4; ++nt)
#pragma unroll
    for (int v = 0; v < 8; ++v) {
      int r = tile * 16 + v + 8 * kh;
      int col = nt * 16 + m;
      float hv = fmaxf(acc[nt][v] + sBn[col], 0.0f);
      h0[r * HD + col] = hv;
      cc[r * HD + col] = hv + sBm[col] + acc2[nt][v];
    }
}

// --- layer: h_new = relu( (deg*h)@W_mh[0:64] + Ah@W_mh[64:128] + c ) --------
__global__ void layer_kernel(const float* __restrict__ h,
                             const float* __restrict__ Ah,
                             const float* __restrict__ deg,
                             const float* __restrict__ cc,
                             const float* __restrict__ W_mh,
                             float* __restrict__ hout) {
  __shared__ __align__(16) float sW[64 * 128];    // 32 KB, [n][k], rows 0..127
  stage_wt<128>(W_mh, sW);
  __syncthreads();

  int lane = threadIdx.x & 31, wave = threadIdx.x >> 5;
  int tile = blockIdx.x * WPB + wave;
  if (tile >= TILES) return;
  int m = lane & 15, kh = lane >> 4;
  int row = tile * 16 + m;
  float dm = deg[row];

  v8f acc[4];
#pragma unroll
  for (int nt = 0; nt < 4; ++nt) acc[nt] = vzero8();

#pragma unroll 4
  for (int ks = 0; ks < 32; ++ks) {
    int k = ks * 4 + kh * 2;
    v2f a;
    if (ks < 16) {
      a = *(const v2f*)&h[row * HD + k];
      a.x *= dm; a.y *= dm;
    } else {
      a = *(const v2f*)&Ah[row * HD + (k - 64)];
    }
#pragma unroll
    for (int nt = 0; nt < 4; ++nt) {
      v2f b = *(const v2f*)&sW[(nt * 16 + m) * 128 + k];
      acc[nt] = wmma4(a, b, acc[nt]);
    }
  }
#pragma unroll
  for (int nt = 0; nt < 4; ++nt)
#pragma unroll
    for (int v = 0; v < 8; ++v) {
      int r = tile * 16 + v + 8 * kh;
      int col = nt * 16 + m;
      hout[r * HD + col] = fmaxf(acc[nt][v] + cc[r * HD + col], 0.0f);
    }
}

// --- zero the pooled accumulator -------------------------------------------
__global__ void zero_pooled_kernel(float* __restrict__ pooled) {
  int t = blockIdx.x * blockDim.x + threadIdx.x;
  if (t < NGR * HD) pooled[t] = 0.0f;
}

// --- out = relu([deg*h, Ah, Ax] @ W_om + b_om); pooled[batch] += out --------
__global__ void final_kernel(const float* __restrict__ h,
                             const float* __restrict__ Ah,
                             const float* __restrict__ Ax,
                             const float* __restrict__ deg,
                             const float* __restrict__ W_om,
                             const float* __restrict__ b_om,
                             const int* __restrict__ batch,
                             float* __restrict__ pooled) {
  __shared__ __align__(16) float sW[64 * 192];    // 48 KB, [n][k]
  __shared__ float sB[HD];
  stage_wt<192>(W_om, sW);
  if (threadIdx.x < HD) sB[threadIdx.x] = b_om[threadIdx.x];
  __syncthreads();

  int lane = threadIdx.x & 31, wave = threadIdx.x >> 5;
  int tile = blockIdx.x * WPB + wave;
  if (tile >= TILES) return;
  int m = lane & 15, kh = lane >> 4;
  int row = tile * 16 + m;
  float dm = deg[row];

  v8f acc[4];
#pragma unroll
  for (int nt = 0; nt < 4; ++nt) acc[nt] = vzero8();

#pragma unroll 4
  for (int ks = 0; ks < 48; ++ks) {
    int k = ks * 4 + kh * 2;
    v2f a;
    if (ks < 16) {
      a = *(const v2f*)&h[row * HD + k];
      a.x *= dm; a.y *= dm;
    } else if (ks < 32) {
      a = *(const v2f*)&Ah[row * HD + (k - 64)];
    } else {
      a = *(const v2f*)&Ax[row * NF + (k - 128)];
    }
#pragma unroll
    for (int nt = 0; nt < 4; ++nt) {
      v2f b = *(const v2f*)&sW[(nt * 16 + m) * 192 + k];
      acc[nt] = wmma4(a, b, acc[nt]);
    }
  }
#pragma unroll
  for (int nt = 0; nt < 4; ++nt)
#pragma unroll
    for (int v = 0; v < 8; ++v) {
      int r = tile * 16 + v + 8 * kh;
      int col = nt * 16 + m;
      float val = fmaxf(acc[nt][v] + sB[col], 0.0f);
      atomicAdd(&pooled[batch[r] * HD + col], val);
    }
}

// --- final GEMV: d_out[g] = pooled[g,:] . W_out + b_out ---------------------
__global__ void pool_out_kernel(const float* __restrict__ pooled,
                                const float* __restrict__ W_out,
                                const float* __restrict__ b_out,
                                float* __restrict__ out) {
  int g = blockIdx.x * blockDim.x + threadIdx.x;
  if (g >= NGR) return;
  float s = b_out[0];
#pragma unroll 8
  for (int f = 0; f < HD; ++f) s += pooled[g * HD + f] * W_out[f];
  out[g] = s;
}

extern "C" void kernel_launch(void* const* d_in, const int* in_sizes, int n_in,
                              void* d_out, int out_size, void* d_ws, size_t ws_size,
                              hipStream_t stream) {
  const float* x     = (const float*)d_in[0];
  const float* ea    = (const float*)d_in[1];
  const float* W_nh  = (const float*)d_in[2];
  const float* b_nh  = (const float*)d_in[3];
  const float* W_mh  = (const float*)d_in[4];
  const float* b_mh  = (const float*)d_in[5];
  const float* W_om  = (const float*)d_in[6];
  const float* b_om  = (const float*)d_in[7];
  const float* W_out = (const float*)d_in[8];
  const float* b_out = (const float*)d_in[9];
  const int*   ei    = (const int*)d_in[10];
  const int*   batch = (const int*)d_in[11];

  float* ws = (float*)d_ws;
  float* hA     = ws; ws += NN * HD;     // h0 / ping
  float* hB     = ws; ws += NN * HD;     // pong
  float* cbuf   = ws; ws += NN * HD;     // layer-invariant additive term
  float* Ah     = ws; ws += NN * HD;     // A @ h  (per layer)
  float* Ax     = ws; ws += NN * NF;     // A @ x  (once)
  float* S_ea   = ws; ws += NN * EF;     // segment_sum of edge_attr (once)
  float* deg    = ws; ws += NN;          // in-degree incl. self loop
  float* pooled = ws; ws += NGR * HD;

  const int nodeThreads = NN * NF;
  const long long edgeThreads = (long long)NE * 64;
  const int edgeBlocks = (int)((edgeThreads + 255) / 256);
  const int gemmBlocks = (TILES + WPB - 1) / WPB;   // 391

  // Precompute: self-loop seed + edge scatter of x / edge_attr / deg
  init_node_kernel<<<(nodeThreads + 255) / 256, 256, 0, stream>>>(x, Ax, S_ea, deg);
  edge_accum_x_kernel<<<edgeBlocks, 256, 0, stream>>>(x, ea, ei, Ax, S_ea, deg);

  // h0 and the layer-invariant term c (WMMA f32)
  h0_c_kernel<<<gemmBlocks, 256, 0, stream>>>(x, W_nh, b_nh, W_mh, b_mh, S_ea, hA, cbuf);

  // 3 message-passing layers
  float* hcur = hA;
  float* hnext = hB;
  for (int L = 0; L < 3; ++L) {
    hipMemcpyAsync(Ah, hcur, (size_t)NN * HD * sizeof(float),
                   hipMemcpyDeviceToDevice, stream);               // self loop seed
    edge_accum_h_kernel<<<edgeBlocks, 256, 0, stream>>>(hcur, ei, Ah);
    layer_kernel<<<gemmBlocks, 256, 0, stream>>>(hcur, Ah, deg, cbuf, W_mh, hnext);
    float* t = hcur; hcur = hnext; hnext = t;
  }

  // Output stage: A@h for final h, fused GEMM+relu+pool, then tiny GEMV
  hipMemcpyAsync(Ah, hcur, (size_t)NN * HD * sizeof(float),
                 hipMemcpyDeviceToDevice, stream);
  edge_accum_h_kernel<<<edgeBlocks, 256, 0, stream>>>(hcur, ei, Ah);
  zero_pooled_kernel<<<(NGR * HD + 255) / 256, 256, 0, stream>>>(pooled);
  final_kernel<<<gemmBlocks, 256, 0, stream>>>(hcur, Ah, Ax, deg, W_om, b_om, batch, pooled);
  pool_out_kernel<<<1, 256, 0, stream>>>(pooled, W_out, b_out, (float*)d_out);
}